// Group_86921548136938
// MI455X (gfx1250) — compile-verified
//
#include <hip/hip_runtime.h>
#include <hip/hip_bf16.h>

// Problem constants (fixed by the reference).
#define BATCH 32
#define NPTS  8192
#define NGRP  512     // G centers per batch
#define KNN   32      // neighbors per center
#define CHUNK 512     // points staged per kNN chunk
#define CTILE 16      // centers per workgroup (one WMMA M-tile)

typedef __attribute__((ext_vector_type(2))) float v2f;
typedef __attribute__((ext_vector_type(4))) float v4f;
typedef __attribute__((ext_vector_type(8))) float v8f;
typedef int v2i_gcc __attribute__((vector_size(8)));   // matches builtin proto

// gfx1250 async global->LDS staging (ASYNCcnt path); guarded so absence of the
// builtin (or the host pass) falls back to plain loads and never breaks compile.
#if defined(__AMDGCN__) && __has_builtin(__builtin_amdgcn_global_load_async_to_lds_b64)
#define USE_ASYNC_LDS 1
#else
#define USE_ASYNC_LDS 0
#endif

__device__ __forceinline__ void wait_async_zero() {
#if defined(__AMDGCN__)
#if __has_builtin(__builtin_amdgcn_s_wait_asynccnt)
    __builtin_amdgcn_s_wait_asynccnt(0);
#else
    asm volatile("s_wait_asynccnt 0" ::: "memory");
#endif
#endif
}

// ---------------------------------------------------------------------------
// Kernel 1: farthest point sampling. One 1024-thread block per batch.
// Each thread owns 8 points (coords + running min-dist) entirely in VGPRs.
// Per round: broadcast last pick, update dists, block-wide argmax
// (wave32 shuffle tree + cross-wave LDS), owner thread publishes next point.
// ---------------------------------------------------------------------------
__global__ __launch_bounds__(1024)
void fps_kernel(const float* __restrict__ xyz, float* __restrict__ centerOut) {
    const int b = blockIdx.x;
    const int t = threadIdx.x;
    const size_t base = (size_t)b * NPTS * 3;

    float px[8], py[8], pz[8], dst[8];
    #pragma unroll
    for (int j = 0; j < 8; ++j) {
        const float* p = xyz + base + (size_t)(t * 8 + j) * 3;
        px[j] = p[0]; py[j] = p[1]; pz[j] = p[2];
        dst[j] = 1.0e10f;
    }

    __shared__ float sRedV[32];
    __shared__ int   sRedI[32];
    __shared__ float sPx, sPy, sPz;
    __shared__ int   sWin;

    if (t == 0) {
        sPx = px[0]; sPy = py[0]; sPz = pz[0];
        float* c0 = centerOut + (size_t)(b * NGRP) * 3;
        c0[0] = px[0]; c0[1] = px[0] == px[0] ? py[0] : py[0]; c0[1] = py[0]; c0[2] = pz[0];
    }
    __syncthreads();

    const int lane = t & 31;
    const int wave = t >> 5;

    for (int g = 1; g < NGRP; ++g) {
        const float lx = sPx, ly = sPy, lz = sPz;

        float best = -1.0f;
        int   bidx = 0x7fffffff;
        #pragma unroll
        for (int j = 0; j < 8; ++j) {
            const float dx = px[j] - lx, dy = py[j] - ly, dz = pz[j] - lz;
            const float d  = dx * dx + dy * dy + dz * dz;
            dst[j] = fminf(dst[j], d);
            if (dst[j] > best) { best = dst[j]; bidx = t * 8 + j; }
        }
        #pragma unroll
        for (int off = 16; off > 0; off >>= 1) {
            const float ov = __shfl_xor(best, off, 32);
            const int   oi = __shfl_xor(bidx, off, 32);
            if (ov > best || (ov == best && oi < bidx)) { best = ov; bidx = oi; }
        }
        if (lane == 0) { sRedV[wave] = best; sRedI[wave] = bidx; }
        __syncthreads();
        if (t == 0) {
            float bv = -1.0f; int bi = 0x7fffffff;
            for (int w = 0; w < 32; ++w) {
                const float v = sRedV[w]; const int i = sRedI[w];
                if (v > bv || (v == bv && i < bi)) { bv = v; bi = i; }
            }
            sWin = bi;
        }
        __syncthreads();
        const int win = sWin;
        if (t == (win >> 3)) {
            const int j = win & 7;
            sPx = px[j]; sPy = py[j]; sPz = pz[j];
            float* c = centerOut + (size_t)(b * NGRP + g) * 3;
            c[0] = px[j]; c[1] = py[j]; c[2] = pz[j];
        }
        __syncthreads();
    }
}

// ---------------------------------------------------------------------------
// Kernel 2: kNN grouping. One 256-thread block per (batch, 16-center tile).
// Distance tiles via V_WMMA_F32_16X16X4_F32:
//   A(m,:) = (-2cx, -2cy, -2cz, |c|^2)   B(:,n) = (x, y, z, 1)
//   C(m,n) = |x_n|^2   =>   D(m,n) = |c_m - x_n|^2
// Point staging uses GLOBAL_LOAD_ASYNC_TO_LDS_B64 when available.
// Fragment builds use ds_load_b128 + register selects (no divergent DS ops).
// ---------------------------------------------------------------------------
__global__ __launch_bounds__(256)
void knn_group_kernel(const float* __restrict__ xyz,
                      const float* __restrict__ centerIn,
                      float* __restrict__ outN) {
    const int gtile = blockIdx.x;
    const int b     = blockIdx.y;
    const int tid   = threadIdx.x;
    const int gbase = gtile * CTILE;

    __shared__ __align__(16) float sPts[CHUNK][4];   // x,y,z,|x|^2   8 KB
    __shared__ __align__(16) float sRaw[CHUNK * 3];  // async stage   6 KB
    __shared__ float sDist[CTILE][CHUNK];            // dist slab    32 KB
    __shared__ float sTopD[CTILE][KNN];
    __shared__ int   sTopI[CTILE][KNN];
    __shared__ float sNewD[CTILE][KNN];
    __shared__ int   sNewI[CTILE][KNN];
    __shared__ float sRdV[CTILE][16];
    __shared__ int   sRdI[CTILE][16];
    __shared__ int   sRdJ[CTILE][16];
    __shared__ float sCx[CTILE], sCy[CTILE], sCz[CTILE], sCn[CTILE];

    if (tid < CTILE) {
        const float* c = centerIn + (size_t)(b * NGRP + gbase + tid) * 3;
        const float cx = c[0], cy = c[1], cz = c[2];
        sCx[tid] = cx; sCy[tid] = cy; sCz[tid] = cz;
        sCn[tid] = cx * cx + cy * cy + cz * cz;
    }
    for (int e = tid; e < CTILE * KNN; e += 256) {
        sTopD[e >> 5][e & 31] = 1.0e30f;
        sTopI[e >> 5][e & 31] = 0;
    }
    __syncthreads();

    const int  lane = tid & 31;
    const int  wv   = tid >> 5;              // 8 waves
    const int  lm   = lane & 15;
    const bool lo   = lane < 16;

    // A fragment: read all components, select in registers (no divergent DS).
    const float acx = sCx[lm], acy = sCy[lm], acz = sCz[lm], acn = sCn[lm];
    v2f A;
    A.x = lo ? (-2.0f * acx) : (-2.0f * acz);   // K=0 / K=2
    A.y = lo ? (-2.0f * acy) : acn;             // K=1 / K=3

    const size_t bN = (size_t)b * NPTS;

    for (int chunk = 0; chunk < NPTS / CHUNK; ++chunk) {
        const int cb = chunk * CHUNK;

        // ---- Stage CHUNK*3 floats (6144 B) of raw xyz into LDS ----
#if USE_ASYNC_LDS
        {
            const char* gsrc = (const char*)xyz + (size_t)(bN + cb) * 12;
            #pragma unroll
            for (int it = 0; it < 3; ++it) {
                const int byteOff = it * 2048 + tid * 8;
                __builtin_amdgcn_global_load_async_to_lds_b64(
                    (__attribute__((address_space(1))) v2i_gcc*)(uintptr_t)(gsrc + byteOff),
                    (__attribute__((address_space(3))) v2i_gcc*)(uint32_t)(uintptr_t)((char*)sRaw + byteOff),
                    0, 0);
            }
            wait_async_zero();
        }
#else
        for (int i = tid; i < CHUNK * 3 / 2; i += 256) {
            const double* g = (const double*)((const char*)xyz + (size_t)(bN + cb) * 12);
            ((double*)sRaw)[i] = g[i];
        }
#endif
        __syncthreads();

        // ---- Build (x,y,z,|x|^2) records with a single b128 store each ----
        for (int i = tid; i < CHUNK; i += 256) {
            const float x = sRaw[i * 3 + 0];
            const float y = sRaw[i * 3 + 1];
            const float z = sRaw[i * 3 + 2];
            v4f r; r.x = x; r.y = y; r.z = z; r.w = x * x + y * y + z * z;
            *(v4f*)&sPts[i][0] = r;
        }
        __syncthreads();

        // ---- 32 tiles of 16 points; wave wv handles tiles wv + 8t ----
        #pragma unroll
        for (int tt = 0; tt < 4; ++tt) {
            const int tb = (wv + tt * 8) * 16;
            const int n  = tb + lm;
            const v4f pv = *(const v4f*)&sPts[n][0];   // ds_load_b128
            v2f Bm;
            Bm.x = lo ? pv.x : pv.z;                   // K=0 / K=2
            Bm.y = lo ? pv.y : 1.0f;                   // K=1 / K=3
            v8f C;
            #pragma unroll
            for (int r = 0; r < 8; ++r) C[r] = pv.w;

            v8f D = __builtin_amdgcn_wmma_f32_16x16x4_f32(
                false, A, false, Bm, (short)0, C, false, false);

            #pragma unroll
            for (int r = 0; r < 8; ++r)
                sDist[r + (lo ? 0 : 8)][n] = D[r];
        }
        __syncthreads();

        // ---- Merge: team m (16 threads) keeps 32 smallest of 512+32 ----
        const int m  = tid >> 4;
        const int tt = tid & 15;
        for (int round = 0; round < KNN; ++round) {
            float bv = 1.0e30f; int bi = 0x7fffffff; int bj = -1;
            for (int j = tt; j < CHUNK + KNN; j += 16) {
                const float v = (j < CHUNK) ? sDist[m][j] : sTopD[m][j - CHUNK];
                const int   i = (j < CHUNK) ? (cb + j)    : sTopI[m][j - CHUNK];
                if (v < bv || (v == bv && i < bi)) { bv = v; bi = i; bj = j; }
            }
            sRdV[m][tt] = bv; sRdI[m][tt] = bi; sRdJ[m][tt] = bj;
            __syncthreads();
            if (tt == 0) {
                float wv2 = 1.0e30f; int wi = 0x7fffffff; int wj = -1;
                for (int s = 0; s < 16; ++s) {
                    const float v = sRdV[m][s]; const int i = sRdI[m][s];
                    if (v < wv2 || (v == wv2 && i < wi)) { wv2 = v; wi = i; wj = sRdJ[m][s]; }
                }
                sNewD[m][round] = wv2; sNewI[m][round] = wi;
                if (wj >= 0) {
                    if (wj < CHUNK) sDist[m][wj] = 1.0e30f;
                    else            sTopD[m][wj - CHUNK] = 1.0e30f;
                }
            }
            __syncthreads();
        }
        for (int e = tid; e < CTILE * KNN; e += 256) {
            sTopD[e >> 5][e & 31] = sNewD[e >> 5][e & 31];
            sTopI[e >> 5][e & 31] = sNewI[e >> 5][e & 31];
        }
        __syncthreads();
    }

    // ---- Gather + recenter ----
    for (int e = tid; e < CTILE * KNN; e += 256) {
        const int m = e >> 5, k = e & 31;
        const int gi = sTopI[m][k];
        const float* p = xyz + (bN + gi) * 3;
        float* o = outN + ((size_t)((b * NGRP + gbase + m) * KNN + k)) * 3;
        o[0] = p[0] - sCx[m];
        o[1] = p[1] - sCy[m];
        o[2] = p[2] - sCz[m];
    }
}

extern "C" void kernel_launch(void* const* d_in, const int* in_sizes, int n_in,
                              void* d_out, int out_size, void* d_ws, size_t ws_size,
                              hipStream_t stream) {
    (void)in_sizes; (void)n_in; (void)d_ws; (void)ws_size; (void)out_size;
    const float* xyz = (const float*)d_in[0];
    float* outAll    = (float*)d_out;
    float* outNbr    = outAll;                                    // [B,G,K,3]
    float* outCtr    = outAll + (size_t)BATCH * NGRP * KNN * 3;   // [B,G,3]

    fps_kernel<<<dim3(BATCH), dim3(1024), 0, stream>>>(xyz, outCtr);
    knn_group_kernel<<<dim3(NGRP / CTILE, BATCH), dim3(256), 0, stream>>>(
        xyz, outCtr, outNbr);
}